// SSIM3D_26714696581756
// MI455X (gfx1250) — compile-verified
//
#include <hip/hip_runtime.h>

typedef __attribute__((ext_vector_type(2))) float v2f;
typedef __attribute__((ext_vector_type(8))) float v8f;

#define N_DIM 160
#define PLANE (N_DIM * N_DIM)        // 25600
#define VOL (N_DIM * N_DIM * N_DIM)  // 4096000
#define NBATCH 4
#define KW 11
#define KRAD 5
#define NCHUNK 7                     // 7 K=4 chunks cover the 26-wide band (28 taps, last 2 zero)
#define TILES_PER_PASS (10 * 10 * N_DIM)  // 16000
#define WAVES_PER_BLOCK 8

// Normalized 11-tap Gaussian, sigma = 1.5 (precomputed offline).
__constant__ float GAUSS11[KW] = {
    0.00102838f, 0.00759876f, 0.03600077f, 0.10936075f, 0.21300580f,
    0.26601354f, 0.21300580f, 0.10936075f, 0.03600077f, 0.00759876f,
    0.00102838f};

// D = A(16x4) * B(4x16) + C, full f32.
__device__ __forceinline__ v8f wmma4(v2f a, v2f b, v8f c) {
    return __builtin_amdgcn_wmma_f32_16x16x4_f32(
        /*neg_a=*/false, a, /*neg_b=*/false, b,
        /*c_mod=*/(short)0, c, /*reuse_a=*/false, /*reuse_b=*/false);
}

// Banded weight matrix W(m, j) = g[j - m] for j-m in [0,10], split into 7 K=4
// chunks. A-matrix 16x4 layout: lanes 0-15 -> M=lane, VGPR0=K0 / VGPR1=K1;
// lanes 16-31 -> M=lane-16, VGPR0=K2 / VGPR1=K3.
__device__ __forceinline__ void make_weights(v2f (&a)[NCHUNK], int lane) {
    int m = lane & 15;
    int koff = (lane >> 4) << 1;  // 0 or 2
#pragma unroll
    for (int c = 0; c < NCHUNK; ++c) {
        int j0 = 4 * c + koff;
        int d0 = j0 - m, d1 = d0 + 1;
        int i0 = min(max(d0, 0), KW - 1);   // clamp -> unconditional table read
        int i1 = min(max(d1, 0), KW - 1);
        float w0 = GAUSS11[i0];
        float w1 = GAUSS11[i1];
        a[c].x = (d0 >= 0 && d0 < KW) ? w0 : 0.f;
        a[c].y = (d1 >= 0 && d1 < KW) ? w1 : 0.f;
    }
}

// ---------- Pass Z: stats on the fly + conv along z -> out[5][VOL] ----------
__global__ __launch_bounds__(256) void ssim_pass_z(
    const float* __restrict__ img1, const float* __restrict__ img2,
    float* __restrict__ out) {
    int lane = threadIdx.x & 31;
    int tile = blockIdx.x * WAVES_PER_BLOCK + (threadIdx.x >> 5);
    v2f a[NCHUNK];
    make_weights(a, lane);
    int y = tile % N_DIM; tile /= N_DIM;
    int xt = tile % 10;   tile /= 10;
    int zt = tile;                       // 0..9
    int x0 = xt * 16, z0 = zt * 16;
    int n = lane & 15;
    int koff = (lane >> 4) << 1;
    int base = y * N_DIM + x0 + n;

    v8f zero = {};
    v8f acc0 = zero, acc1 = zero, acc2 = zero, acc3 = zero, acc4 = zero;
#pragma unroll
    for (int c = 0; c < NCHUNK; ++c) {
        int j = 4 * c + koff;
        int za = z0 + j - KRAD;
        int zb = za + 1;
        bool ia = (za >= 0) && (za < N_DIM);
        bool ib = (zb >= 0) && (zb < N_DIM);
        int oa = min(max(za, 0), N_DIM - 1) * PLANE + base;  // clamped, always valid
        int ob = min(max(zb, 0), N_DIM - 1) * PLANE + base;
        float i1a = img1[oa];            // unconditional loads
        float i2a = img2[oa];
        float i1b = img1[ob];
        float i2b = img2[ob];
        i1a = ia ? i1a : 0.f;            // zero-padding via cndmask, no branches
        i2a = ia ? i2a : 0.f;
        i1b = ib ? i1b : 0.f;
        i2b = ib ? i2b : 0.f;
        v2f b1  = {i1a, i1b};
        v2f b2  = {i2a, i2b};
        v2f b11 = {i1a * i1a, i1b * i1b};
        v2f b22 = {i2a * i2a, i2b * i2b};
        v2f b12 = {i1a * i2a, i1b * i2b};
        acc0 = wmma4(a[c], b1,  acc0);
        acc1 = wmma4(a[c], b2,  acc1);
        acc2 = wmma4(a[c], b11, acc2);
        acc3 = wmma4(a[c], b22, acc3);
        acc4 = wmma4(a[c], b12, acc4);
    }
    int mbase = (lane >> 4) << 3;  // 0 or 8
#pragma unroll
    for (int r = 0; r < 8; ++r) {
        int o = (z0 + mbase + r) * PLANE + base;
        out[0 * VOL + o] = acc0[r];
        out[1 * VOL + o] = acc1[r];
        out[2 * VOL + o] = acc2[r];
        out[3 * VOL + o] = acc3[r];
        out[4 * VOL + o] = acc4[r];
    }
}

// ---------- Pass Y: conv along y for 5 stat volumes ----------
__global__ __launch_bounds__(256) void ssim_pass_y(
    const float* __restrict__ in, float* __restrict__ out) {
    int lane = threadIdx.x & 31;
    int tile = blockIdx.x * WAVES_PER_BLOCK + (threadIdx.x >> 5);
    v2f a[NCHUNK];
    make_weights(a, lane);
    int z = tile % N_DIM; tile /= N_DIM;
    int xt = tile % 10;   tile /= 10;
    int yt = tile;
    int x0 = xt * 16, y0 = yt * 16;
    int n = lane & 15;
    int koff = (lane >> 4) << 1;
    int base = z * PLANE + x0 + n;

    v8f zero = {};
    v8f acc[5];
#pragma unroll
    for (int s = 0; s < 5; ++s) acc[s] = zero;
#pragma unroll
    for (int c = 0; c < NCHUNK; ++c) {
        int j = 4 * c + koff;
        int ya = y0 + j - KRAD;
        int yb = ya + 1;
        bool ba = (ya >= 0) && (ya < N_DIM);
        bool bb = (yb >= 0) && (yb < N_DIM);
        int oa = min(max(ya, 0), N_DIM - 1) * N_DIM + base;
        int ob = min(max(yb, 0), N_DIM - 1) * N_DIM + base;
#pragma unroll
        for (int s = 0; s < 5; ++s) {
            float va = in[s * VOL + oa];
            float vb = in[s * VOL + ob];
            va = ba ? va : 0.f;
            vb = bb ? vb : 0.f;
            v2f bv = {va, vb};
            acc[s] = wmma4(a[c], bv, acc[s]);
        }
    }
    int mbase = (lane >> 4) << 3;
#pragma unroll
    for (int r = 0; r < 8; ++r) {
        int o = (y0 + mbase + r) * N_DIM + base;
#pragma unroll
        for (int s = 0; s < 5; ++s) out[s * VOL + o] = acc[s][r];
    }
}

// ---------- Pass X: conv along x + SSIM map + reduction ----------
__global__ __launch_bounds__(256) void ssim_pass_x(
    const float* __restrict__ in, float* __restrict__ accum) {
    int lane = threadIdx.x & 31;
    int tile = blockIdx.x * WAVES_PER_BLOCK + (threadIdx.x >> 5);
    v2f a[NCHUNK];
    make_weights(a, lane);
    int z = tile % N_DIM; tile /= N_DIM;
    int yt = tile % 10;   tile /= 10;
    int xt = tile;
    int x0 = xt * 16, y0 = yt * 16;
    int n = lane & 15;              // N = y line
    int koff = (lane >> 4) << 1;
    int base = z * PLANE + (y0 + n) * N_DIM;

    v8f zero = {};
    v8f acc[5];
#pragma unroll
    for (int s = 0; s < 5; ++s) acc[s] = zero;
#pragma unroll
    for (int c = 0; c < NCHUNK; ++c) {
        int j = 4 * c + koff;
        int xa = x0 + j - KRAD;
        int xb = xa + 1;
        bool ba = (xa >= 0) && (xa < N_DIM);
        bool bb = (xb >= 0) && (xb < N_DIM);
        int oa = base + min(max(xa, 0), N_DIM - 1);
        int ob = base + min(max(xb, 0), N_DIM - 1);
#pragma unroll
        for (int s = 0; s < 5; ++s) {
            float va = in[s * VOL + oa];
            float vb = in[s * VOL + ob];
            va = ba ? va : 0.f;
            vb = bb ? vb : 0.f;
            v2f bv = {va, vb};
            acc[s] = wmma4(a[c], bv, acc[s]);
        }
    }
    const float C1 = 0.0001f;  // 0.01^2
    const float C2 = 0.0009f;  // 0.03^2
    float sum = 0.f;
#pragma unroll
    for (int r = 0; r < 8; ++r) {
        float mu1 = acc[0][r], mu2 = acc[1][r];
        float e11 = acc[2][r], e22 = acc[3][r], e12 = acc[4][r];
        float mu1s = mu1 * mu1, mu2s = mu2 * mu2, m12 = mu1 * mu2;
        float s1 = e11 - mu1s, s2 = e22 - mu2s, s12 = e12 - m12;
        float num = (2.f * m12 + C1) * (2.f * s12 + C2);
        float den = (mu1s + mu2s + C1) * (s1 + s2 + C2);
        sum += num / den;
    }
    // wave32 reduction
#pragma unroll
    for (int off = 16; off >= 1; off >>= 1) sum += __shfl_xor(sum, off);
    if (lane == 0) atomicAdd(accum, sum);
}

__global__ void ssim_init_acc(float* acc) { acc[0] = 0.f; }

__global__ void ssim_finalize(const float* __restrict__ acc, float* __restrict__ out) {
    out[0] = acc[0] * (1.0f / (float)(NBATCH * (long)VOL));
}

extern "C" void kernel_launch(void* const* d_in, const int* in_sizes, int n_in,
                              void* d_out, int out_size, void* d_ws, size_t ws_size,
                              hipStream_t stream) {
    (void)in_sizes; (void)n_in; (void)out_size; (void)ws_size;
    const float* img1 = (const float*)d_in[0];
    const float* img2 = (const float*)d_in[1];
    float* out = (float*)d_out;

    char* ws = (char*)d_ws;
    float* acc  = (float*)ws;                  // 1 float accumulator
    float* buf0 = (float*)(ws + 256);          // 5 * VOL floats
    float* buf1 = buf0 + 5L * VOL;             // 5 * VOL floats

    const int grid = TILES_PER_PASS / WAVES_PER_BLOCK;  // 2000 blocks of 8 waves

    ssim_init_acc<<<1, 1, 0, stream>>>(acc);
    for (int b = 0; b < NBATCH; ++b) {
        const float* i1 = img1 + (long)b * VOL;
        const float* i2 = img2 + (long)b * VOL;
        ssim_pass_z<<<grid, 256, 0, stream>>>(i1, i2, buf0);
        ssim_pass_y<<<grid, 256, 0, stream>>>(buf0, buf1);
        ssim_pass_x<<<grid, 256, 0, stream>>>(buf1, acc);
    }
    ssim_finalize<<<1, 1, 0, stream>>>(acc, out);
}